// OrthoConvV2_37495064494812
// MI455X (gfx1250) — compile-verified
//
#include <hip/hip_runtime.h>
#include <math.h>
#include <stdint.h>

// ---------------------------------------------------------------------------
// OrthoConvV2 on MI455X (gfx1250, wave32).
// Pipeline per layer: build V -> direct 9-term VFFT -> per-frequency
// Gauss-Jordan orthogonalization W = I - 2 V (V^H V)^-1 V^H -> fused rfft2
// (DFT matmuls in LDS) -> per-frequency complex GEMM on WMMA f32 16x16x4
// with async-to-LDS staging of W (Karatsuba, 3 wmma per tile-step)
// -> fused irfft2 -> mean/bias/ReLU.
// ---------------------------------------------------------------------------

#define WIDTH 64
#define RANK  32
#define BATCH 128
#define DEPTH 6
#define NMAX  76
#define MMAX  39
#define NMMAX (NMAX * MMAX)          // 2964 frequencies max per layer
#define SCALE 0.40824829046386302f   // 6^-0.5
#define PI2   6.28318530717958647f

typedef __attribute__((ext_vector_type(2))) float v2f;
typedef __attribute__((ext_vector_type(8))) float v8f;
typedef __attribute__((ext_vector_type(4))) int   v4i;
typedef __attribute__((address_space(1))) v4i     as1_v4i;
typedef __attribute__((address_space(3))) v4i     as3_v4i;

// --------------------------- async copy helpers ----------------------------
// CDNA5 GLOBAL_LOAD_ASYNC_TO_LDS_B128 (ASYNCcnt-tracked LDS DMA).
#if defined(__has_builtin)
#if __has_builtin(__builtin_amdgcn_global_load_async_to_lds_b128)
#define HAVE_ASYNC_BUILTIN 1
#endif
#if __has_builtin(__builtin_amdgcn_s_wait_asynccnt)
#define HAVE_ASYNCCNT_BUILTIN 1
#endif
#endif

__device__ __forceinline__ void async_copy_b128(const float* g, float* l) {
#if defined(HAVE_ASYNC_BUILTIN)
  __builtin_amdgcn_global_load_async_to_lds_b128(
      (as1_v4i*)g, (as3_v4i*)l, 0, 0);
#else
  unsigned lds_off = (unsigned)(uintptr_t)(as3_v4i*)l;
  asm volatile("global_load_async_to_lds_b128 %0, %1, off"
               :: "v"(lds_off), "v"(g) : "memory");
#endif
}

__device__ __forceinline__ void async_wait0() {
#if defined(HAVE_ASYNCCNT_BUILTIN)
  __builtin_amdgcn_s_wait_asynccnt(0);
#else
  asm volatile("s_wait_asynccnt 0" ::: "memory");
#endif
}

// ---------------------------------------------------------------------------
// 1) V = identity_kernel + weights * gamma * SCALE         [D,W,R,3,3]
// ---------------------------------------------------------------------------
__global__ void k_build_V(const float* __restrict__ w,
                          const float* __restrict__ g,
                          float* __restrict__ V) {
  int idx = blockIdx.x * blockDim.x + threadIdx.x;
  const int total = DEPTH * WIDTH * RANK * 9;
  if (idx >= total) return;
  int kk = idx % 9;
  int ky = kk / 3, kx = kk % 3;
  int r  = (idx / 9) % RANK;
  int wc = (idx / (9 * RANK)) % WIDTH;
  int d  = idx / (9 * RANK * WIDTH);
  float ident = (wc < RANK && r == wc && ky == 1 && kx == 1) ? 1.0f : 0.0f;
  V[idx] = ident + w[idx] * g[d * WIDTH + wc] * SCALE;
}

// ---------------------------------------------------------------------------
// 2) VFFT[f][w][r] = shift(u,v) * conj(rfft2(V, s=(n,n), ortho))
//    3x3 support -> direct 9-term sum. One thread per (f,w,r).
// ---------------------------------------------------------------------------
__global__ void k_vfft(const float* __restrict__ Vl,   // layer base [W,R,3,3]
                       float* __restrict__ VFr, float* __restrict__ VFi,
                       int n, int m) {
  int tid = blockIdx.x * blockDim.x + threadIdx.x;
  int nm = n * m;
  if (tid >= nm * WIDTH * RANK) return;
  int f  = tid / (WIDTH * RANK);
  int rest = tid % (WIDTH * RANK);
  int wc = rest / RANK;
  int r  = rest % RANK;
  int u = f / m, v = f % m;
  float ang0 = PI2 / (float)n;
  float ar = 0.f, ai = 0.f;
  // conj(rfft2) sum: (1/n) * sum V * e^{+2pi i (u ky + v kx)/n}
#pragma unroll
  for (int ky = 0; ky < 3; ++ky) {
#pragma unroll
    for (int kx = 0; kx < 3; ++kx) {
      float val = Vl[(wc * RANK + r) * 9 + ky * 3 + kx];
      float ph = ang0 * (float)(u * ky + v * kx);
      float s, c;
      __sincosf(ph, &s, &c);
      ar += val * c;
      ai += val * s;
    }
  }
  // shift = e^{-2pi i (u+v)/n}; scale 1/n (ortho)
  float ps = -ang0 * (float)(u + v);
  float ss, cs;
  __sincosf(ps, &ss, &cs);
  float inv_n = 1.0f / (float)n;
  VFr[(size_t)f * (WIDTH * RANK) + rest] = (ar * cs - ai * ss) * inv_n;
  VFi[(size_t)f * (WIDTH * RANK) + rest] = (ar * ss + ai * cs) * inv_n;
}

// ---------------------------------------------------------------------------
// 3) Per-frequency: G = V^H V + eps I ; solve G M = V^H (Gauss-Jordan) ;
//    W = I - 2 V M.  One workgroup (256 thr) per frequency, all in LDS.
// ---------------------------------------------------------------------------
__global__ void k_build_W(const float* __restrict__ VFr,
                          const float* __restrict__ VFi,
                          float* __restrict__ Wr, float* __restrict__ Wi) {
  __shared__ float sVr[WIDTH][RANK], sVi[WIDTH][RANK];       // 16 KB
  __shared__ float sGr[RANK][RANK + 1], sGi[RANK][RANK + 1]; // 8.25 KB
  __shared__ float sMr[RANK][WIDTH], sMi[RANK][WIDTH];       // 16 KB
  __shared__ float f_r[RANK], f_i[RANK];
  __shared__ float piv[2];

  const int f = blockIdx.x;
  const int tid = threadIdx.x;
  const float* vr = VFr + (size_t)f * (WIDTH * RANK);
  const float* vi = VFi + (size_t)f * (WIDTH * RANK);

  for (int e = tid; e < WIDTH * RANK; e += 256) {
    sVr[e / RANK][e % RANK] = vr[e];
    sVi[e / RANK][e % RANK] = vi[e];
  }
  __syncthreads();

  // Gram: G[r][s] = sum_w conj(V[w][r]) V[w][s]
  for (int e = tid; e < RANK * RANK; e += 256) {
    int r = e / RANK, s = e % RANK;
    float gr = 0.f, gi = 0.f;
    for (int w = 0; w < WIDTH; ++w) {
      float cr = sVr[w][r], ci = -sVi[w][r];
      float br = sVr[w][s], bi = sVi[w][s];
      gr += cr * br - ci * bi;
      gi += cr * bi + ci * br;
    }
    sGr[r][s] = gr;
    sGi[r][s] = gi;
  }
  __syncthreads();

  // eps = mean(diag) * 1e-7 added to diagonal
  if (tid == 0) {
    float tr = 0.f;
    for (int r = 0; r < RANK; ++r) tr += sGr[r][r];
    float eps = (tr / (float)RANK) * 1e-7f;
    for (int r = 0; r < RANK; ++r) sGr[r][r] += eps;
  }
  // RHS: M = V^H   (32 x 64)
  for (int e = tid; e < RANK * WIDTH; e += 256) {
    int r = e / WIDTH, w = e % WIDTH;
    sMr[r][w] = sVr[w][r];
    sMi[r][w] = -sVi[w][r];
  }
  __syncthreads();

  // Gauss-Jordan on [G | M]
  for (int k = 0; k < RANK; ++k) {
    if (tid == 0) {
      float pr = sGr[k][k], pi = sGi[k][k];
      float d = pr * pr + pi * pi + 1e-30f;
      piv[0] = pr / d;
      piv[1] = -pi / d;
    }
    __syncthreads();
    float ir = piv[0], ii = piv[1];
    for (int e = tid; e < RANK + WIDTH; e += 256) {
      if (e < RANK) {
        float xr = sGr[k][e], xi = sGi[k][e];
        sGr[k][e] = xr * ir - xi * ii;
        sGi[k][e] = xr * ii + xi * ir;
      } else {
        int w = e - RANK;
        float xr = sMr[k][w], xi = sMi[k][w];
        sMr[k][w] = xr * ir - xi * ii;
        sMi[k][w] = xr * ii + xi * ir;
      }
    }
    __syncthreads();
    if (tid < RANK) {
      f_r[tid] = sGr[tid][k];
      f_i[tid] = sGi[tid][k];
    }
    __syncthreads();
    for (int e = tid; e < RANK * (RANK + WIDTH); e += 256) {
      int i = e / (RANK + WIDTH);
      int c = e % (RANK + WIDTH);
      if (i == k) continue;
      float fr = f_r[i], fi = f_i[i];
      if (c < RANK) {
        float kr = sGr[k][c], ki = sGi[k][c];
        sGr[i][c] -= fr * kr - fi * ki;
        sGi[i][c] -= fr * ki + fi * kr;
      } else {
        int w = c - RANK;
        float kr = sMr[k][w], ki = sMi[k][w];
        sMr[i][w] -= fr * kr - fi * ki;
        sMi[i][w] -= fr * ki + fi * kr;
      }
    }
    __syncthreads();
  }

  // W = I - 2 V M
  float* wrp = Wr + (size_t)f * (WIDTH * WIDTH);
  float* wip = Wi + (size_t)f * (WIDTH * WIDTH);
  for (int e = tid; e < WIDTH * WIDTH; e += 256) {
    int p = e / WIDTH, q = e % WIDTH;
    float ar = 0.f, ai = 0.f;
    for (int r = 0; r < RANK; ++r) {
      float vrp = sVr[p][r], vip = sVi[p][r];
      float mr = sMr[r][q], mi = sMi[r][q];
      ar += vrp * mr - vip * mi;
      ai += vrp * mi + vip * mr;
    }
    wrp[e] = ((p == q) ? 1.0f : 0.0f) - 2.0f * ar;
    wip[e] = -2.0f * ai;
  }
}

// ---------------------------------------------------------------------------
// 4) Fused rfft2 (ortho) of zero-padded image, one workgroup per (b,w) image.
//    DFT as two matmul passes in LDS with incremental twiddle rotation.
//    Output layout: X[(u*m+v)][w][b]  (K-major for the WMMA GEMM).
// ---------------------------------------------------------------------------
__global__ void k_rfft2(const float* __restrict__ x, int s,
                        float* __restrict__ Xr, float* __restrict__ Xi,
                        int n, int m) {
  __shared__ float xp[NMAX][NMAX + 1];            // padded image
  __shared__ float Tr[16][NMAX + 1], Ti[16][NMAX + 1];

  const int img = blockIdx.x;            // b*WIDTH + w
  const int b = img / WIDTH, w = img % WIDTH;
  const int tid = threadIdx.x;
  const float* src = x + (size_t)img * s * s;
  const float ang0 = PI2 / (float)n;
  const float inv_n = 1.0f / (float)n;

  for (int e = tid; e < n * n; e += 256) {
    int i = e / n, j = e % n;
    float v = 0.f;
    if (i >= 1 && i <= s && j >= 1 && j <= s) v = src[(i - 1) * s + (j - 1)];
    xp[i][j] = v;
  }
  __syncthreads();

  for (int u0 = 0; u0 < n; u0 += 16) {
    // stage A: T[u][x2] = sum_x1 xp[x1][x2] e^{-2pi i u x1 / n}
    for (int e = tid; e < 16 * n; e += 256) {
      int uu = e / n, x2 = e % n;
      int u = u0 + uu;
      if (u < n) {
        float dth = -ang0 * (float)u;
        float ds, dc;
        __sincosf(dth, &ds, &dc);
        float c = 1.f, sn = 0.f, sr = 0.f, si = 0.f;
        for (int x1 = 0; x1 < n; ++x1) {
          float val = xp[x1][x2];
          sr += val * c;
          si += val * sn;
          float t = c * dc - sn * ds;
          sn = c * ds + sn * dc;
          c = t;
        }
        Tr[uu][x2] = sr;
        Ti[uu][x2] = si;
      }
    }
    __syncthreads();
    // stage B: X[u][v] = (1/n) sum_x2 T[u][x2] e^{-2pi i v x2 / n}
    for (int e = tid; e < 16 * m; e += 256) {
      int uu = e / m, v = e % m;
      int u = u0 + uu;
      if (u < n) {
        float dth = -ang0 * (float)v;
        float ds, dc;
        __sincosf(dth, &ds, &dc);
        float c = 1.f, sn = 0.f, ar = 0.f, ai = 0.f;
        for (int x2 = 0; x2 < n; ++x2) {
          float tr = Tr[uu][x2], ti = Ti[uu][x2];
          ar += tr * c - ti * sn;
          ai += tr * sn + ti * c;
          float t = c * dc - sn * ds;
          sn = c * ds + sn * dc;
          c = t;
        }
        size_t o = ((size_t)(u * m + v)) * (WIDTH * BATCH) + w * BATCH + b;
        Xr[o] = ar * inv_n;
        Xi[o] = ai * inv_n;
      }
    }
    __syncthreads();
  }
}

// ---------------------------------------------------------------------------
// 5) Per-frequency complex GEMM on WMMA:  Z[f] = W[f] @ X[f]
//    W: 64x64 complex (staged LDS via async DMA), X: 64x128 complex.
//    One workgroup (8 waves) per f; wave (rt,ch) owns 16 rows x 64 cols.
//    Karatsuba: P1=Wr*Xr, P2=Wi*Xi, P3=(Wr+Wi)(Xr+Xi);
//               Zr = P1-P2, Zi = P3-P1-P2.   3 wmma per (kstep, ntile).
// ---------------------------------------------------------------------------
__global__ void k_freq_gemm(const float* __restrict__ Wr,
                            const float* __restrict__ Wi,
                            const float* __restrict__ Xr,
                            const float* __restrict__ Xi,
                            float* __restrict__ Zr,
                            float* __restrict__ Zi) {
  __shared__ __align__(16) float sW[2][WIDTH][WIDTH];   // [0]=re [1]=im, 32 KB

  const int f = blockIdx.x;
  const float* wr = Wr + (size_t)f * (WIDTH * WIDTH);
  const float* wi = Wi + (size_t)f * (WIDTH * WIDTH);
  const float* xr = Xr + (size_t)f * (WIDTH * BATCH);
  const float* xi = Xi + (size_t)f * (WIDTH * BATCH);
  float* zr = Zr + (size_t)f * (WIDTH * BATCH);
  float* zi = Zi + (size_t)f * (WIDTH * BATCH);

  __builtin_prefetch(xr, 0, 1);   // global_prefetch_b8: pull X panel toward L2
  __builtin_prefetch(xi, 0, 1);

  const int tid = threadIdx.x;
  // Stage W (re+im) into LDS: 2048 x b128 chunks, 8 per thread, ASYNCcnt path.
  for (int e = tid; e < (WIDTH * WIDTH) / 4; e += 256) {
    async_copy_b128(wr + e * 4, &sW[0][0][0] + e * 4);
    async_copy_b128(wi + e * 4, &sW[1][0][0] + e * 4);
  }
  async_wait0();
  __syncthreads();

  const int lane = tid & 31;
  const int wid = tid >> 5;
  const int rt = wid & 3;          // row tile   (4 x 16 rows)
  const int ch = wid >> 2;         // col half   (2 x 64 cols)
  const int lo = lane & 15;
  const int khalf = lane >> 4;     // A frag: lanes 0-15 hold K=0,1; 16-31 K=2,3
  const int mrow = rt * 16 + lo;

  v8f p1[4], p2[4], p3[4];
#pragma unroll
  for (int t = 0; t < 4; ++t) { p1[t] = (v8f)0.f; p2[t] = (v8f)0.f; p3[t] = (v8f)0.f; }

#pragma unroll 2
  for (int k0 = 0; k0 < WIDTH; k0 += 4) {
    const int ka = k0 + 2 * khalf;               // even -> 8B-aligned LDS reads
    v2f a_r = *(const v2f*)&sW[0][mrow][ka];     // ds_load_b64
    v2f a_i = *(const v2f*)&sW[1][mrow][ka];
    v2f a_s = a_r + a_i;
#pragma unroll
    for (int t = 0; t < 4; ++t) {
      const int col = ch * 64 + t * 16 + lo;
      v2f b_r, b_i, b_s;
      b_r[0] = xr[(ka + 0) * BATCH + col];
      b_r[1] = xr[(ka + 1) * BATCH + col];
      b_i[0] = xi[(ka + 0) * BATCH + col];
      b_i[1] = xi[(ka + 1) * BATCH + col];
      b_s = b_r + b_i;
      p1[t] = __builtin_amdgcn_wmma_f32_16x16x4_f32(false, a_r, false, b_r,
                                                    (short)0, p1[t], false, false);
      p2[t] = __builtin_amdgcn_wmma_f32_16x16x4_f32(false, a_i, false, b_i,
                                                    (short)0, p2[t], false, false);
      p3[t] = __builtin_amdgcn_wmma_f32_16x16x4_f32(false, a_s, false, b_s,
                                                    (short)0, p3[t], false, false);
    }
  }

  // C/D layout: VGPR v -> row = 16*rt + v + 8*khalf, col = 64*ch + 16*t + lo
#pragma unroll
  for (int t = 0; t < 4; ++t) {
    v8f cr = p1[t] - p2[t];
    v8f ci = p3[t] - p1[t] - p2[t];
    const int col = ch * 64 + t * 16 + lo;
#pragma unroll
    for (int v = 0; v < 8; ++v) {
      int row = rt * 16 + v + 8 * khalf;
      zr[row * BATCH + col] = cr[v];
      zi[row * BATCH + col] = ci[v];
    }
  }
}

// ---------------------------------------------------------------------------
// 6) Fused irfft2 (ortho) per image using hermitian half-spectrum.
// ---------------------------------------------------------------------------
__global__ void k_irfft2(const float* __restrict__ Zr,
                         const float* __restrict__ Zi,
                         float* __restrict__ xout, int n, int m) {
  __shared__ float sZr[NMAX][MMAX + 1], sZi[NMAX][MMAX + 1];
  __shared__ float sSr[16][MMAX + 1], sSi[16][MMAX + 1];

  const int img = blockIdx.x;
  const int b = img / WIDTH, w = img % WIDTH;
  const int tid = threadIdx.x;
  const float ang0 = PI2 / (float)n;
  const float inv_n = 1.0f / (float)n;

  for (int e = tid; e < n * m; e += 256) {
    int u = e / m, v = e % m;
    size_t o = ((size_t)e) * (WIDTH * BATCH) + w * BATCH + b;
    sZr[u][v] = Zr[o];
    sZi[u][v] = Zi[o];
  }
  __syncthreads();

  float* dst = xout + (size_t)img * n * n;
  for (int x10 = 0; x10 < n; x10 += 16) {
    // stage A: S[x1][v] = sum_u Z[u][v] e^{+2pi i u x1/n}
    for (int e = tid; e < 16 * m; e += 256) {
      int xx = e / m, v = e % m;
      int x1 = x10 + xx;
      if (x1 < n) {
        float dth = ang0 * (float)x1;
        float ds, dc;
        __sincosf(dth, &ds, &dc);
        float c = 1.f, sn = 0.f, ar = 0.f, ai = 0.f;
        for (int u = 0; u < n; ++u) {
          float zrv = sZr[u][v], ziv = sZi[u][v];
          ar += zrv * c - ziv * sn;
          ai += zrv * sn + ziv * c;
          float t = c * dc - sn * ds;
          sn = c * ds + sn * dc;
          c = t;
        }
        sSr[xx][v] = ar;
        sSi[xx][v] = ai;
      }
    }
    __syncthreads();
    // stage B: z[x1][x2] = (1/n) sum_v alpha_v Re(S[x1][v] e^{+2pi i v x2/n})
    for (int e = tid; e < 16 * n; e += 256) {
      int xx = e / n, x2 = e % n;
      int x1 = x10 + xx;
      if (x1 < n) {
        float dth = ang0 * (float)x2;
        float ds, dc;
        __sincosf(dth, &ds, &dc);
        float c = 1.f, sn = 0.f, acc = 0.f;
        for (int v = 0; v < m; ++v) {
          float alpha = (v == 0 || v == m - 1) ? 1.0f : 2.0f;  // n even, Nyquist
          acc += alpha * (sSr[xx][v] * c - sSi[xx][v] * sn);
          float t = c * dc - sn * ds;
          sn = c * ds + sn * dc;
          c = t;
        }
        dst[x1 * n + x2] = acc * inv_n;
      }
    }
    __syncthreads();
  }
}

// ---------------------------------------------------------------------------
// 7) Per-channel mean over (batch, spatial); deterministic block reduction.
// ---------------------------------------------------------------------------
__global__ void k_chan_mean(const float* __restrict__ xin,
                            float* __restrict__ mean, int n) {
  __shared__ float red[256];
  const int w = blockIdx.x;
  const int tid = threadIdx.x;
  const int sp = n * n;
  float acc = 0.f;
  for (int e = tid; e < BATCH * sp; e += 256) {
    int b = e / sp, o = e % sp;
    acc += xin[((size_t)(b * WIDTH + w)) * sp + o];
  }
  red[tid] = acc;
  __syncthreads();
  for (int s = 128; s > 0; s >>= 1) {
    if (tid < s) red[tid] += red[tid + s];
    __syncthreads();
  }
  if (tid == 0) mean[w] = red[0] / (float)(BATCH * sp);
}

// ---------------------------------------------------------------------------
// 8) x = relu(z + bias - mean)
// ---------------------------------------------------------------------------
__global__ void k_bias_relu(const float* __restrict__ xin,
                            float* __restrict__ xout,
                            const float* __restrict__ bias,
                            const float* __restrict__ mean, int n) {
  size_t idx = (size_t)blockIdx.x * blockDim.x + threadIdx.x;
  size_t total = (size_t)BATCH * WIDTH * n * n;
  if (idx >= total) return;
  int w = (int)((idx / (size_t)(n * n)) % WIDTH);
  float v = xin[idx] + bias[w] - mean[w];
  xout[idx] = v > 0.f ? v : 0.f;
}

// ---------------------------------------------------------------------------
// Host launcher
// ---------------------------------------------------------------------------
extern "C" void kernel_launch(void* const* d_in, const int* in_sizes, int n_in,
                              void* d_out, int out_size, void* d_ws, size_t ws_size,
                              hipStream_t stream) {
  const float* d_x = (const float*)d_in[0];       // [128,64,64,64]
  const float* d_wt = (const float*)d_in[1];      // [6,64,32,3,3]
  const float* d_gm = (const float*)d_in[2];      // [6,64,1,1,1]
  const float* d_bs = (const float*)d_in[3];      // [6,64]

  float* ws = (float*)d_ws;
  const size_t VSZ = (size_t)DEPTH * WIDTH * RANK * 9;
  const size_t VFSZ = (size_t)NMMAX * WIDTH * RANK;
  const size_t WSZ = (size_t)NMMAX * WIDTH * WIDTH;
  const size_t XSZ = (size_t)NMMAX * WIDTH * BATCH;
  const size_t IMSZ = (size_t)BATCH * WIDTH * NMAX * NMAX;

  float* pV = ws;
  float* pVFr = pV + VSZ;
  float* pVFi = pVFr + VFSZ;
  float* pWr = pVFi + VFSZ;
  float* pWi = pWr + WSZ;
  float* pXr = pWi + WSZ;
  float* pXi = pXr + XSZ;
  float* pZr = pXi + XSZ;
  float* pZi = pZr + XSZ;
  float* pxA = pZi + XSZ;
  float* pxB = pxA + IMSZ;
  float* pMean = pxB + IMSZ;

  k_build_V<<<(unsigned)((VSZ + 255) / 256), 256, 0, stream>>>(d_wt, d_gm, pV);

  const int ns[DEPTH] = {66, 68, 70, 72, 74, 76};
  const float* xsrc = d_x;
  for (int i = 0; i < DEPTH; ++i) {
    const int n = ns[i];
    const int m = n / 2 + 1;
    const int nm = n * m;
    const int s = n - 2;
    float* xdst = (i & 1) ? pxB : pxA;

    k_vfft<<<(unsigned)(((size_t)nm * WIDTH * RANK + 255) / 256), 256, 0, stream>>>(
        pV + (size_t)i * WIDTH * RANK * 9, pVFr, pVFi, n, m);
    k_build_W<<<nm, 256, 0, stream>>>(pVFr, pVFi, pWr, pWi);
    k_rfft2<<<BATCH * WIDTH, 256, 0, stream>>>(xsrc, s, pXr, pXi, n, m);
    k_freq_gemm<<<nm, 256, 0, stream>>>(pWr, pWi, pXr, pXi, pZr, pZi);
    k_irfft2<<<BATCH * WIDTH, 256, 0, stream>>>(pZr, pZi, xdst, n, m);
    k_chan_mean<<<WIDTH, 256, 0, stream>>>(xdst, pMean, n);

    float* finaldst = (i == DEPTH - 1) ? (float*)d_out : xdst;
    size_t total = (size_t)BATCH * WIDTH * n * n;
    k_bias_relu<<<(unsigned)((total + 255) / 256), 256, 0, stream>>>(
        xdst, finaldst, d_bs + (size_t)i * WIDTH, pMean, n);

    xsrc = xdst;
  }
}